// EEGGNN_6863357739128
// MI455X (gfx1250) — compile-verified
//
#include <hip/hip_runtime.h>

typedef __attribute__((ext_vector_type(2))) float v2f;
typedef __attribute__((ext_vector_type(8))) float v8f;

#define D_FEAT 96
#define HIDDEN 32
#define WPB 8   // waves per block in the MLP kernel

// ---------------------------------------------------------------------------
// 1) agg = x  (GIN eps=0: h = agg + x folded into initial value); zero pools
// ---------------------------------------------------------------------------
__global__ void init_kernel(const float* __restrict__ x, float* __restrict__ agg,
                            float* __restrict__ sums, float* __restrict__ counts,
                            int total, int nGraphs) {
  int i = blockIdx.x * blockDim.x + threadIdx.x;
  if (i < total) agg[i] = x[i];
  if (i < nGraphs * HIDDEN) sums[i] = 0.0f;
  if (i < nGraphs) counts[i] = 0.0f;
}

// ---------------------------------------------------------------------------
// 2) scatter-add over edges: agg[dst] += x[src].  One wave per edge; each of
//    the 32 lanes handles 3 of the 96 features -> global_atomic_add_f32.
// ---------------------------------------------------------------------------
__global__ void scatter_kernel(const float* __restrict__ x, const int* __restrict__ ei,
                               float* __restrict__ agg, int nEdges) {
  int gid = blockIdx.x * blockDim.x + threadIdx.x;
  int e = gid >> 5;
  int lane = gid & 31;
  if (e >= nEdges) return;
  int s = ei[e];            // src row
  int d = ei[nEdges + e];   // dst row
  const float* xs = x + (size_t)s * D_FEAT;
  float* ad = agg + (size_t)d * D_FEAT;
#pragma unroll
  for (int f = lane; f < D_FEAT; f += 32)
    atomicAdd(&ad[f], xs[f]);
}

// ---------------------------------------------------------------------------
// 3) per-node MLP: h2 = relu(relu(h @ W1 + b1) @ W2 + b2) using
//    V_WMMA_F32_16X16X4_F32.  One wave computes a 16-node tile.
//
//    A fragment (16x4 f32, 2 VGPR): lane<16 -> row=lane, K={k0,k0+1};
//                                   lane>=16 -> row=lane-16, K={k0+2,k0+3}.
//    B fragment (4x16 f32, 2 VGPR): mirrored, col = lane&15.
//    C/D (16x16 f32, 8 VGPR): VGPR i -> M = i + (lane>=16 ? 8 : 0), N = lane&15.
// ---------------------------------------------------------------------------
__global__ void __launch_bounds__(256) mlp_kernel(
    const float* __restrict__ h, const float* __restrict__ W1,
    const float* __restrict__ b1, const float* __restrict__ W2,
    const float* __restrict__ b2, float* __restrict__ out,
    int nTiles, int nNodes) {
  __shared__ float sW1[D_FEAT * HIDDEN];     // 12 KB
  __shared__ float sW2[HIDDEN * HIDDEN];     // 4 KB
  __shared__ float sB1[HIDDEN];
  __shared__ float sB2[HIDDEN];
  __shared__ float sT[WPB][16 * HIDDEN];     // per-wave 16x32 staging (16 KB)

  const int tid = threadIdx.x;
  for (int i = tid; i < D_FEAT * HIDDEN; i += blockDim.x) sW1[i] = W1[i];
  for (int i = tid; i < HIDDEN * HIDDEN; i += blockDim.x) sW2[i] = W2[i];
  if (tid < HIDDEN) { sB1[tid] = b1[tid]; sB2[tid] = b2[tid]; }
  __syncthreads();

  const int wid  = tid >> 5;
  const int lane = tid & 31;
  const int tile = blockIdx.x * WPB + wid;
  if (tile >= nTiles) return;   // wave-uniform; barriers already done

  const int row = lane & 15;    // M (for A) / N (for B,C,D)
  const int kh  = lane >> 4;    // which K-half this lane holds
  const int col = lane & 15;

  int node = tile * 16 + row;
  int nclamp = node < nNodes ? node : (nNodes - 1);
  const float* hrow = h + (size_t)nclamp * D_FEAT;

  // ---- GEMM1: [16x96] @ [96x32] ----
  v8f acc0 = {}; v8f acc1 = {};
#pragma unroll
  for (int k0 = 0; k0 < D_FEAT; k0 += 4) {
    const int k = k0 + 2 * kh;
    v2f a;  a.x  = hrow[k];                      a.y  = hrow[k + 1];
    v2f bA; bA.x = sW1[k * HIDDEN + col];        bA.y = sW1[(k + 1) * HIDDEN + col];
    v2f bB; bB.x = sW1[k * HIDDEN + col + 16];   bB.y = sW1[(k + 1) * HIDDEN + col + 16];
    acc0 = __builtin_amdgcn_wmma_f32_16x16x4_f32(false, a, false, bA, (short)0, acc0, false, false);
    acc1 = __builtin_amdgcn_wmma_f32_16x16x4_f32(false, a, false, bB, (short)0, acc1, false, false);
  }

  // bias + ReLU, stage C/D layout -> LDS (row-major 16x32) for re-use as A
  {
    const float bias0 = sB1[col], bias1 = sB1[col + 16];
    float* tw = sT[wid];
#pragma unroll
    for (int i = 0; i < 8; ++i) {
      const int m = i + kh * 8;
      float v0 = acc0[i] + bias0; v0 = v0 > 0.f ? v0 : 0.f;
      float v1 = acc1[i] + bias1; v1 = v1 > 0.f ? v1 : 0.f;
      tw[m * HIDDEN + col]      = v0;
      tw[m * HIDDEN + col + 16] = v1;
    }
  }

  // ---- GEMM2: [16x32] @ [32x32] ----
  const float* tw = sT[wid];
  v8f c0 = {}; v8f c1 = {};
#pragma unroll
  for (int k0 = 0; k0 < HIDDEN; k0 += 4) {
    const int k = k0 + 2 * kh;
    v2f a;  a.x  = tw[row * HIDDEN + k];         a.y  = tw[row * HIDDEN + k + 1];
    v2f bA; bA.x = sW2[k * HIDDEN + col];        bA.y = sW2[(k + 1) * HIDDEN + col];
    v2f bB; bB.x = sW2[k * HIDDEN + col + 16];   bB.y = sW2[(k + 1) * HIDDEN + col + 16];
    c0 = __builtin_amdgcn_wmma_f32_16x16x4_f32(false, a, false, bA, (short)0, c0, false, false);
    c1 = __builtin_amdgcn_wmma_f32_16x16x4_f32(false, a, false, bB, (short)0, c1, false, false);
  }

  // bias + outer ReLU, store h2
  {
    const float bb0 = sB2[col], bb1 = sB2[col + 16];
#pragma unroll
    for (int i = 0; i < 8; ++i) {
      const int m = i + kh * 8;
      const int n = tile * 16 + m;
      if (n < nNodes) {
        float v0 = c0[i] + bb0; v0 = v0 > 0.f ? v0 : 0.f;
        float v1 = c1[i] + bb1; v1 = v1 > 0.f ? v1 : 0.f;
        out[(size_t)n * HIDDEN + col]      = v0;
        out[(size_t)n * HIDDEN + col + 16] = v1;
      }
    }
  }
}

// ---------------------------------------------------------------------------
// 4) pooled sums: wave per node, lane per feature
// ---------------------------------------------------------------------------
__global__ void pool_kernel(const float* __restrict__ h2, const int* __restrict__ batch,
                            float* __restrict__ sums, float* __restrict__ counts,
                            int nNodes) {
  int gid = blockIdx.x * blockDim.x + threadIdx.x;
  int n = gid >> 5;
  int f = gid & 31;
  if (n >= nNodes) return;
  int g = batch[n];
  atomicAdd(&sums[g * HIDDEN + f], h2[(size_t)n * HIDDEN + f]);
  if (f == 0) atomicAdd(&counts[g], 1.0f);
}

// ---------------------------------------------------------------------------
// 5) head: out[g] = (sums[g]/max(count,1)) @ Wc + bc
// ---------------------------------------------------------------------------
__global__ void head_kernel(const float* __restrict__ sums, const float* __restrict__ counts,
                            const float* __restrict__ Wc, const float* __restrict__ bc,
                            float* __restrict__ out, int nGraphs) {
  int g = blockIdx.x * blockDim.x + threadIdx.x;
  if (g >= nGraphs) return;
  float c = counts[g]; c = c > 1.0f ? c : 1.0f;
  float inv = 1.0f / c;
  float o0 = bc[0], o1 = bc[1];
#pragma unroll
  for (int f = 0; f < HIDDEN; ++f) {
    float p = sums[g * HIDDEN + f] * inv;
    o0 += p * Wc[f * 2 + 0];
    o1 += p * Wc[f * 2 + 1];
  }
  out[g * 2 + 0] = o0;
  out[g * 2 + 1] = o1;
}

// ---------------------------------------------------------------------------
extern "C" void kernel_launch(void* const* d_in, const int* in_sizes, int n_in,
                              void* d_out, int out_size, void* d_ws, size_t ws_size,
                              hipStream_t stream) {
  const float* x     = (const float*)d_in[0];
  const int*   ei    = (const int*)d_in[1];
  const int*   batch = (const int*)d_in[2];
  const float* W1    = (const float*)d_in[3];
  const float* b1    = (const float*)d_in[4];
  const float* W2    = (const float*)d_in[5];
  const float* b2    = (const float*)d_in[6];
  const float* Wc    = (const float*)d_in[7];
  const float* bc    = (const float*)d_in[8];
  float* out = (float*)d_out;

  const int nNodes  = in_sizes[0] / D_FEAT;   // 50000
  const int nEdges  = in_sizes[1] / 2;        // 800000
  const int nGraphs = out_size / 2;           // 256

  // workspace layout (floats)
  float* agg    = (float*)d_ws;                       // nNodes*96
  float* h2     = agg + (size_t)nNodes * D_FEAT;      // nNodes*32
  float* sums   = h2 + (size_t)nNodes * HIDDEN;       // nGraphs*32
  float* counts = sums + (size_t)nGraphs * HIDDEN;    // nGraphs

  // 1) init
  {
    int total = nNodes * D_FEAT;
    int blocks = (total + 255) / 256;
    init_kernel<<<blocks, 256, 0, stream>>>(x, agg, sums, counts, total, nGraphs);
  }
  // 2) scatter-add over edges
  {
    long long tthreads = (long long)nEdges * 32;
    int blocks = (int)((tthreads + 255) / 256);
    scatter_kernel<<<blocks, 256, 0, stream>>>(x, ei, agg, nEdges);
  }
  // 3) node MLP with WMMA
  {
    int nTiles = (nNodes + 15) / 16;          // 3125 (exact)
    int blocks = (nTiles + WPB - 1) / WPB;
    mlp_kernel<<<blocks, 256, 0, stream>>>(agg, W1, b1, W2, b2, h2, nTiles, nNodes);
  }
  // 4) pooled sums
  {
    long long tthreads = (long long)nNodes * 32;
    int blocks = (int)((tthreads + 255) / 256);
    pool_kernel<<<blocks, 256, 0, stream>>>(h2, batch, sums, counts, nNodes);
  }
  // 5) head
  {
    int blocks = (nGraphs + 255) / 256;
    head_kernel<<<blocks, 256, 0, stream>>>(sums, counts, Wc, bc, out, nGraphs);
  }
}